// CenterLoss_67207648247984
// MI455X (gfx1250) — compile-verified
//
#include <hip/hip_runtime.h>

// Center loss: mean_b clamp(||x_b - centers[labels_b]||^2, 1e-12, 1e12)
// B=2048, C=100000, D=128, all f32; labels int64.
//
// Strategy: each wave32 handles 16 rows. Row reduction done via
// V_WMMA_F32_16X16X4_F32 with A = (x-c)^2 tile (16x4) and B = ones (4x16):
// D[i][j] = sum_k A[i][k] for all j, accumulated over 32 K-steps (D=128).
// Deterministic two-stage reduction (block partials -> single-thread sum).

typedef float v2f __attribute__((ext_vector_type(2)));
typedef float v8f __attribute__((ext_vector_type(8)));

#define BATCH 2048
#define DIM 128
#define ROWS_PER_WAVE 16
#define WAVES_PER_BLOCK 4
#define ROWS_PER_BLOCK (ROWS_PER_WAVE * WAVES_PER_BLOCK)   // 64
#define NUM_BLOCKS (BATCH / ROWS_PER_BLOCK)                // 32
#define CLAMP_MIN 1e-12f
#define CLAMP_MAX 1e12f

__global__ void center_loss_partial(const float* __restrict__ x,
                                    const float* __restrict__ centers,
                                    const long long* __restrict__ labels,
                                    float* __restrict__ partials) {
  const int lane  = threadIdx.x & 31;
  const int wave  = threadIdx.x >> 5;                    // 0..3
  const int m     = lane & 15;                           // row within wave tile
  const int khalf = lane >> 4;                           // 0: K=0,1  1: K=2,3
  const int row   = blockIdx.x * ROWS_PER_BLOCK + wave * ROWS_PER_WAVE + m;

  const long long lbl = labels[row];
  const float* __restrict__ xr = x       + (size_t)row * DIM;
  const float* __restrict__ cr = centers + (size_t)lbl * DIM;

  v2f bOnes;
  bOnes.x = 1.0f;
  bOnes.y = 1.0f;

  v8f acc = {};
#pragma unroll
  for (int kk = 0; kk < DIM / 4; ++kk) {
    const int col = kk * 4 + khalf * 2;                  // 8B-aligned
    const v2f xv = *(const v2f*)(xr + col);
    const v2f cv = *(const v2f*)(cr + col);
    const float d0 = xv.x - cv.x;
    const float d1 = xv.y - cv.y;
    v2f a;
    a.x = d0 * d0;
    a.y = d1 * d1;
    // D = A(16x4 of squared diffs) x B(4x16 ones) + C  -> every column of D
    // holds the row sums; chain C to accumulate over the D dimension.
    acc = __builtin_amdgcn_wmma_f32_16x16x4_f32(
        /*neg_a=*/false, a, /*neg_b=*/false, bOnes,
        /*c_mod=*/(short)0, acc, /*reuse_a=*/false, /*reuse_b=*/false);
  }

  // Lanes 0-15: acc[r] = dist(row r of tile), r=0..7.
  // Lanes 16-31: acc[r] = dist(row r+8).  Clamp per-row, then sum.
  float s = 0.0f;
#pragma unroll
  for (int i = 0; i < 8; ++i) {
    float d = acc[i];
    d = fminf(fmaxf(d, CLAMP_MIN), CLAMP_MAX);
    s += d;
  }
  // Merge the two half-wave partials (identical across lanes within each half).
  float waveSum = s + __shfl_xor(s, 16, 32);

  __shared__ float lds[WAVES_PER_BLOCK];
  if (lane == 0) lds[wave] = waveSum;
  __syncthreads();
  if (threadIdx.x == 0) {
    float p = 0.0f;
#pragma unroll
    for (int w = 0; w < WAVES_PER_BLOCK; ++w) p += lds[w];
    partials[blockIdx.x] = p;
  }
}

__global__ void center_loss_finalize(const float* __restrict__ partials,
                                     float* __restrict__ out) {
  if (threadIdx.x == 0 && blockIdx.x == 0) {
    float s = 0.0f;
#pragma unroll
    for (int i = 0; i < NUM_BLOCKS; ++i) s += partials[i];
    out[0] = s * (1.0f / (float)BATCH);
  }
}

extern "C" void kernel_launch(void* const* d_in, const int* in_sizes, int n_in,
                              void* d_out, int out_size, void* d_ws, size_t ws_size,
                              hipStream_t stream) {
  const float*     x       = (const float*)d_in[0];
  const float*     centers = (const float*)d_in[1];
  const long long* labels  = (const long long*)d_in[2];
  float* out      = (float*)d_out;
  float* partials = (float*)d_ws;   // NUM_BLOCKS floats of scratch

  center_loss_partial<<<NUM_BLOCKS, WAVES_PER_BLOCK * 32, 0, stream>>>(
      x, centers, labels, partials);
  center_loss_finalize<<<1, 32, 0, stream>>>(partials, out);
}